// AttentionModule_81277961109667
// MI455X (gfx1250) — compile-verified
//
#include <hip/hip_runtime.h>
#include <hip/hip_bf16.h>
#include <math.h>

// Problem constants (fixed by reference setup_inputs)
#define BB 4
#define NN 10
#define KK 5
#define QQ 512
#define DD 1536
#define NKR 50          // N*K rows
#define QC 8            // query chunks for reduction parallelism
#define RQT 384         // reduce_q threads: 384 * float4 = 1536 = D

typedef __attribute__((ext_vector_type(2))) float v2f;
typedef __attribute__((ext_vector_type(4))) float v4f;
typedef __attribute__((ext_vector_type(8))) float v8f;

// ---------------------------------------------------------------------------
// Kernel 1: partial reduction of query -> column sums (per q-chunk) + sum(q^2)
// grid = B*QC blocks, 384 threads; each thread owns 4 consecutive d's so a
// block streams each 6144-byte query row with contiguous b128 loads.
// ---------------------------------------------------------------------------
__global__ void reduce_q_kernel(const float* __restrict__ query,
                                float* __restrict__ qpart,   // [B][QC][D]
                                float* __restrict__ sqpart)  // [B][QC]
{
    __shared__ float red[RQT];
    const int tid = threadIdx.x;
    const int qc  = blockIdx.x % QC;
    const int b   = blockIdx.x / QC;

    const int d  = tid * 4;
    const int q0 = qc * (QQ / QC);
    const float* qp = query + ((size_t)b * QQ + q0) * DD + d;

    v4f acc = {};
    float accsq = 0.f;
    #pragma unroll 4
    for (int q = 0; q < QQ / QC; ++q) {
        v4f v = *(const v4f*)(qp + (size_t)q * DD);
        acc += v;
        accsq += v.x * v.x + v.y * v.y + v.z * v.z + v.w * v.w;
    }
    *(v4f*)(qpart + ((size_t)b * QC + qc) * DD + d) = acc;

    // block-reduce accsq over 384 threads (fold 384 -> 128 -> tree)
    red[tid] = accsq;
    __syncthreads();
    if (tid < 128) red[tid] += red[tid + 128] + red[tid + 256];
    __syncthreads();
    for (int s = 64; s > 0; s >>= 1) {
        if (tid < s) red[tid] += red[tid + s];
        __syncthreads();
    }
    if (tid == 0) sqpart[b * QC + qc] = red[0];
}

// ---------------------------------------------------------------------------
// Kernel 2: fold partials -> Qbar[b][d] = mean_q query, meanqsq[b] = mean||q||^2
// grid = B blocks, 256 threads
// ---------------------------------------------------------------------------
__global__ void finalize_q_kernel(const float* __restrict__ qpart,
                                  const float* __restrict__ sqpart,
                                  float* __restrict__ qbar,     // [B][D]
                                  float* __restrict__ meanqsq)  // [B]
{
    __shared__ float red[8];
    const int tid = threadIdx.x;
    const int b   = blockIdx.x;
    const float inv = 1.0f / (float)QQ;

    #pragma unroll
    for (int j = 0; j < DD / 256; ++j) {
        const int d = j * 256 + tid;
        float s = 0.f;
        #pragma unroll
        for (int qc = 0; qc < QC; ++qc)
            s += qpart[((size_t)b * QC + qc) * DD + d];
        qbar[(size_t)b * DD + d] = s * inv;
    }

    if (tid < 8) red[tid] = sqpart[b * QC + tid];
    __syncthreads();
    if (tid == 0) {
        float s = 0.f;
        #pragma unroll
        for (int i = 0; i < 8; ++i) s += red[i];
        meanqsq[b] = s * inv;
    }
}

// ---------------------------------------------------------------------------
// Kernel 3: per batch b:
//   raw[s] = S_s.S_s - 2 S_s.Qbar  via diag of WMMA f32 16x16x4 tile chains
//   logit  = tanh(-(raw + meanqsq)); softmax over K; weighted agg over shots
// grid = B blocks, 256 threads (8 waves; waves 0..3 drive WMMA, one tile each)
// ---------------------------------------------------------------------------
__global__ void proto_attn_kernel(const float* __restrict__ support, // [B][NK][D]
                                  const float* __restrict__ qbar,    // [B][D]
                                  const float* __restrict__ meanqsq, // [B]
                                  float* __restrict__ agg_out,       // [B][N][D]
                                  float* __restrict__ qgw_out)       // [B][N][K]
{
    __shared__ float raw[64];   // tile-padded NK rows
    __shared__ float wgt[NN * KK];
    __shared__ float mq_s;

    const int tid  = threadIdx.x;
    const int lane = tid & 31;
    const int wave = tid >> 5;
    const int b    = blockIdx.x;

    if (wave < 4) {
        // One 16-row tile of S per wave. f32 WMMA lane layout:
        //  A 16x4: row = lane%16, k-pair base = 4j + 2*(lane/16)
        //  B 4x16: col = lane%16, same k-pair base
        // => both operands come from the SAME float2 of S per lane.
        //
        // Rows >= NKR are clamped to row 0 (in-bounds). No masking needed:
        // a bogus A-row m only pollutes C row m, a bogus B-col n only
        // pollutes C col n, and we only ever read diag[m] for valid m < NKR,
        // which depends solely on the valid A-row m and valid B-col m.
        const int tile  = wave;
        const int r     = lane & 15;
        const int khalf = (lane >> 4) << 1; // 0 or 2
        const int srow  = tile * 16 + r;
        const int srow_c = (srow < NKR) ? srow : 0;

        const float* Sp = support + ((size_t)b * NKR + srow_c) * DD;
        const float* Qp = qbar + (size_t)b * DD;

        v8f c0 = {}; v8f c1 = {};
        #pragma unroll 4
        for (int j = 0; j < DD / 4; j += 2) {
            const int k0 = j * 4 + khalf;
            const int k1 = k0 + 4;
            v2f a0 = *(const v2f*)(Sp + k0);
            v2f a1 = *(const v2f*)(Sp + k1);
            v2f q0 = *(const v2f*)(Qp + k0);
            v2f q1 = *(const v2f*)(Qp + k1);
            v2f b0 = a0 - 2.f * q0;   // lowers to v_pk_fma_f32
            v2f b1 = a1 - 2.f * q1;
            c0 = __builtin_amdgcn_wmma_f32_16x16x4_f32(
                     false, a0, false, b0, (short)0, c0, false, false);
            c1 = __builtin_amdgcn_wmma_f32_16x16x4_f32(
                     false, a1, false, b1, (short)0, c1, false, false);
        }
        v8f c = c0 + c1;

        // Extract diagonal of the 16x16 C tile:
        //   m in [0,8):  VGPR m,   lane m
        //   m in [8,16): VGPR m-8, lane m+16
        #pragma unroll
        for (int v = 0; v < 8; ++v) {
            if (lane == v)            raw[tile * 16 + v]     = c[v];
            else if (lane == v + 24)  raw[tile * 16 + v + 8] = c[v];
        }
    }
    if (tid == 0) mq_s = meanqsq[b];
    __syncthreads();

    // tanh + softmax over K within each class n (one thread per n)
    if (tid < NN) {
        const int n = tid;
        float l[KK], m = -1e30f;
        #pragma unroll
        for (int k = 0; k < KK; ++k) {
            l[k] = tanhf(-(raw[n * KK + k] + mq_s));
            m = fmaxf(m, l[k]);
        }
        float e[KK], s = 0.f;
        #pragma unroll
        for (int k = 0; k < KK; ++k) { e[k] = __expf(l[k] - m); s += e[k]; }
        const float invs = 1.0f / s;
        #pragma unroll
        for (int k = 0; k < KK; ++k) {
            const float w = e[k] * invs;
            wgt[n * KK + k] = w;
            qgw_out[(b * NN + n) * KK + k] = w;
        }
    }
    __syncthreads();

    // Weighted aggregation over shots: agg[b,n,d] = sum_k S[b,n,k,d] * w[n,k]
    #pragma unroll
    for (int n = 0; n < NN; ++n) {
        const float w0 = wgt[n * KK + 0];
        const float w1 = wgt[n * KK + 1];
        const float w2 = wgt[n * KK + 2];
        const float w3 = wgt[n * KK + 3];
        const float w4 = wgt[n * KK + 4];
        const float* sp = support + (((size_t)b * NN + n) * KK) * DD;
        float* op = agg_out + ((size_t)b * NN + n) * DD;
        for (int d = tid; d < DD; d += 256) {
            float acc = sp[d] * w0
                      + sp[d + DD] * w1
                      + sp[d + 2 * DD] * w2
                      + sp[d + 3 * DD] * w3
                      + sp[d + 4 * DD] * w4;
            op[d] = acc;
        }
    }
}

// ---------------------------------------------------------------------------
extern "C" void kernel_launch(void* const* d_in, const int* in_sizes, int n_in,
                              void* d_out, int out_size, void* d_ws, size_t ws_size,
                              hipStream_t stream)
{
    const float* support = (const float*)d_in[0];
    const float* query   = (const float*)d_in[1];
    float* out = (float*)d_out;
    float* ws  = (float*)d_ws;

    // workspace layout (floats)
    float* qpart   = ws;                         // B*QC*D = 49152
    float* sqpart  = ws + (size_t)BB * QC * DD;  // B*QC   = 32
    float* qbar    = sqpart + 64;                // B*D    = 6144 (padded start)
    float* meanqsq = qbar + (size_t)BB * DD;     // B      = 4

    float* agg_out = out;                        // B*N*D = 61440
    float* qgw_out = out + (size_t)BB * NN * DD; // B*N*K = 200

    reduce_q_kernel<<<BB * QC, RQT, 0, stream>>>(query, qpart, sqpart);
    finalize_q_kernel<<<BB, 256, 0, stream>>>(qpart, sqpart, qbar, meanqsq);
    proto_attn_kernel<<<BB, 256, 0, stream>>>(support, qbar, meanqsq, agg_out, qgw_out);
}